// Matchmaker_vNext_v1_42674795053926
// MI455X (gfx1250) — compile-verified
//
#include <hip/hip_runtime.h>
#include <math.h>

typedef float v2f __attribute__((ext_vector_type(2)));
typedef float v8f __attribute__((ext_vector_type(8)));

#define DEVFN static __device__ __forceinline__

DEVFN float wred_sum(float v) {
  #pragma unroll
  for (int off = 16; off; off >>= 1) v += __shfl_xor(v, off, 32);
  return v;
}
DEVFN float wred_max(float v) {
  #pragma unroll
  for (int off = 16; off; off >>= 1) v = fmaxf(v, __shfl_xor(v, off, 32));
  return v;
}

DEVFN v8f wmma4(v2f a, v2f b, v8f c) {
  return __builtin_amdgcn_wmma_f32_16x16x4_f32(false, a, false, b,
                                               (short)0, c, false, false);
}

// ---------------------------------------------------------------------------
// Positional feature add: out = x + pos(T, 300)
// ---------------------------------------------------------------------------
__global__ void posadd_kernel(const float* __restrict__ x, float* __restrict__ out,
                              int T, int total) {
  int idx = blockIdx.x * blockDim.x + threadIdx.x;
  if (idx >= total) return;
  int e = idx % 300;
  int t = (idx / 300) % T;
  const float c = 9.2103403719761836f / 149.0f; // log(10000)/(nt-1), nt=150
  int j = (e < 150) ? e : (e - 150);
  float inv = expf(-c * (float)j);
  float ang = (float)t * inv;
  float pf = (e < 150) ? sinf(ang) : cosf(ang);
  out[idx] = x[idx] + pf;
}

// ---------------------------------------------------------------------------
// K-loop helper: two WMMA accumulator chains, K unrolled by 8, K % 4 == 0.
// MASKED=false -> full tile, no fragment scaling (common case, shortest path).
// TRANSB: B accessed as B[n][k] (ldb = K-stride), else B[k][n].
// ---------------------------------------------------------------------------
template <bool MASKED, bool TRANSB>
DEVFN v8f kloop(const float* __restrict__ pa, const float* __restrict__ pb,
                int ldb, int K, float mscale, float nscale) {
  v8f acc0 = {}, acc1 = {};
  int K8 = K & ~7;
  if (TRANSB) {
    for (int kb = 0; kb < K8; kb += 8) {
      __builtin_prefetch(pa + 64, 0, 1);
      __builtin_prefetch(pb + 64, 0, 1);
      v2f a0 = *(const v2f*)(pa);
      v2f a1 = *(const v2f*)(pa + 4);
      v2f b0 = *(const v2f*)(pb);
      v2f b1 = *(const v2f*)(pb + 4);
      if (MASKED) {
        a0 *= mscale; a1 *= mscale;
        b0 *= nscale; b1 *= nscale;
      }
      acc0 = wmma4(a0, b0, acc0);
      acc1 = wmma4(a1, b1, acc1);
      pa += 8; pb += 8;
    }
    if (K8 < K) { // single 4-wide tail (K % 8 == 4)
      v2f a0 = *(const v2f*)(pa);
      v2f b0 = *(const v2f*)(pb);
      if (MASKED) { a0 *= mscale; b0 *= nscale; }
      acc0 = wmma4(a0, b0, acc0);
    }
  } else {
    int ldb4 = 4 * ldb;
    for (int kb = 0; kb < K8; kb += 8) {
      __builtin_prefetch(pa + 64, 0, 1);
      v2f a0 = *(const v2f*)(pa);
      v2f a1 = *(const v2f*)(pa + 4);
      v2f b0, b1;
      b0.x = pb[0];
      b0.y = pb[ldb];
      b1.x = pb[ldb4];
      b1.y = pb[ldb4 + ldb];
      if (MASKED) {
        a0 *= mscale; a1 *= mscale;
        b0 *= nscale; b1 *= nscale;
      }
      acc0 = wmma4(a0, b0, acc0);
      acc1 = wmma4(a1, b1, acc1);
      pa += 8; pb += 2 * ldb4;
    }
    if (K8 < K) {
      v2f a0 = *(const v2f*)(pa);
      v2f b0;
      b0.x = pb[0];
      b0.y = pb[ldb];
      if (MASKED) { a0 *= mscale; b0 *= nscale; }
      acc0 = wmma4(a0, b0, acc0);
    }
  }
  return acc0 + acc1;
}

// ---------------------------------------------------------------------------
// Generic fp32 WMMA GEMM: C[b] = act( A[b] @ B[b] + bias [, / norm product] )
// One wave per 16x16 tile. K % 4 == 0 required (call sites: 300/100/64).
// Full tiles take the unmasked fast path (wave-uniform branch, EXEC stays ~0).
// act: 0 none, 1 relu, 2 tanh. rsn/csn: x /= ((rsn[m]+1e-13)*(csn[n]+1e-13))
// ---------------------------------------------------------------------------
__global__ void gemm_wmma_f32(const float* __restrict__ A, int lda, int strideA,
                              const float* __restrict__ Bm, int ldb, int strideB,
                              int transB,
                              const float* __restrict__ bias,
                              float* __restrict__ C, int ldc, int strideC,
                              int M, int N, int K, int act,
                              const float* __restrict__ rsn, int rs_stride,
                              const float* __restrict__ csn, int cs_stride) {
  int lane = threadIdx.x & 31;
  int n0 = blockIdx.x * 16;
  int m0 = blockIdx.y * 16;
  int b  = blockIdx.z;
  const float* Ab = A  + (long)b * strideA;
  const float* Bb = Bm + (long)b * strideB;
  float*       Cb = C  + (long)b * strideC;

  int r    = lane & 15;   // A row in tile / B,C column in tile
  int half = lane >> 4;   // K-half for A/B fragments, M-half for C/D
  int mrow = m0 + r;
  int ncol = n0 + r;
  bool full = (m0 + 16 <= M) && (n0 + 16 <= N);  // wave-uniform
  float mscale = (mrow < M) ? 1.0f : 0.0f;
  float nscale = (ncol < N) ? 1.0f : 0.0f;
  int arow = (mrow < M) ? mrow : (M - 1);
  int bcol = (ncol < N) ? ncol : (N - 1);

  // 8-byte aligned in all call sites (lda/ldb even, 2*half even)
  const float* pa  = Ab + arow * lda + 2 * half;
  const float* pbT = Bb + bcol * ldb + 2 * half;        // transB: B[n][k]
  const float* pbN = Bb + (2 * half) * ldb + bcol;      // normal: B[k][n]

  v8f acc;
  if (transB) {
    acc = full ? kloop<false, true>(pa, pbT, ldb, K, mscale, nscale)
               : kloop<true,  true>(pa, pbT, ldb, K, mscale, nscale);
  } else {
    acc = full ? kloop<false, false>(pa, pbN, ldb, K, mscale, nscale)
               : kloop<true,  false>(pa, pbN, ldb, K, mscale, nscale);
  }

  float bval = bias ? bias[bcol] : 0.0f;
  float csv = 1.0f;
  if (rsn) csv = csn[b * cs_stride + bcol] + 1e-13f;
  #pragma unroll
  for (int v = 0; v < 8; ++v) {
    int m = m0 + v + 8 * half;  // C layout: VGPR v holds rows v and v+8
    int n = ncol;
    if (m < M && n < N) {
      float x = acc[v] + bval;
      if (rsn) {
        float rv = rsn[b * rs_stride + m] + 1e-13f;
        x = x / (rv * csv);
      }
      if (act == 1) x = fmaxf(x, 0.0f);
      else if (act == 2) x = tanhf(x);
      Cb[m * ldc + n] = x;
    }
  }
}

// ---------------------------------------------------------------------------
// Residual + LayerNorm over E=300: out = (g*(a+b - mean)/(std+1e-6) + beta)*mask
// One wave per row, 4 waves per block.
// ---------------------------------------------------------------------------
__global__ void ln_kernel(const float* __restrict__ ina, const float* __restrict__ inb,
                          const float* __restrict__ g, const float* __restrict__ be,
                          const float* __restrict__ rowmask,
                          float* __restrict__ out, int rows) {
  const int E = 300;
  int wave = threadIdx.x >> 5, lane = threadIdx.x & 31;
  int row = blockIdx.x * 4 + wave;
  if (row >= rows) return;
  const float* pa = ina + row * E;
  const float* pb = inb + row * E;
  float vals[10];
  float sum = 0.0f;
  #pragma unroll
  for (int t = 0; t < 10; ++t) {
    int idx = lane + t * 32;
    float v = (idx < E) ? (pa[idx] + pb[idx]) : 0.0f;
    vals[t] = v;
    sum += v;
  }
  sum = wred_sum(sum);
  float mean = sum / 300.0f;
  float var = 0.0f;
  #pragma unroll
  for (int t = 0; t < 10; ++t) {
    int idx = lane + t * 32;
    float d = (idx < E) ? (vals[t] - mean) : 0.0f;
    var += d * d;
  }
  var = wred_sum(var);
  float inv = 1.0f / (sqrtf(var / 300.0f) + 1e-6f);
  float mk = rowmask ? rowmask[row] : 1.0f;
  float* po = out + row * E;
  #pragma unroll
  for (int t = 0; t < 10; ++t) {
    int idx = lane + t * 32;
    if (idx < E) po[idx] = (g[idx] * (vals[t] - mean) * inv + be[idx]) * mk;
  }
}

// ---------------------------------------------------------------------------
// MHSA with H=32 heads, head dim 2, scale = sqrt(E//H) = 3.
// comb layout per token: [q(64) | k(64) | v(64)]. One wave per query token.
// Exactly replicates: p = softmax(sim*m)*m; p /= (sum p + 1e-13); o = p@v
// ---------------------------------------------------------------------------
__global__ void attn_kernel(const float* __restrict__ comb, const float* __restrict__ mask,
                            float* __restrict__ atto, int T) {
  __shared__ float smask[512];
  int b = blockIdx.y;
  int wave = threadIdx.x >> 5, lane = threadIdx.x & 31;
  int i = blockIdx.x * 4 + wave;
  for (int j = threadIdx.x; j < T; j += blockDim.x) smask[j] = mask[b * T + j];
  __syncthreads();

  const float* rowq = comb + (long)(b * T + i) * 192;
  const float inv3 = 1.0f / 3.0f;
  int chunks = T >> 5;
  for (int h = 0; h < 32; ++h) {
    float q0 = rowq[2 * h] * inv3;
    float q1 = rowq[2 * h + 1] * inv3;
    // pass 1: row max of t_j = sim_j * m_j
    float lmax = -3.4e38f;
    for (int c = 0; c < chunks; ++c) {
      int j = c * 32 + lane;
      const float* rk = comb + (long)(b * T + j) * 192 + 64;
      float s = (q0 * rk[2 * h] + q1 * rk[2 * h + 1]) * smask[j];
      lmax = fmaxf(lmax, s);
    }
    lmax = wred_max(lmax);
    // pass 2: sums + weighted v accumulation
    float Sall = 0.0f, Smask = 0.0f, o0 = 0.0f, o1 = 0.0f;
    for (int c = 0; c < chunks; ++c) {
      int j = c * 32 + lane;
      const float* rk = comb + (long)(b * T + j) * 192 + 64;
      float mj = smask[j];
      float s = (q0 * rk[2 * h] + q1 * rk[2 * h + 1]) * mj;
      float e = expf(s - lmax);
      float me = e * mj;
      Sall += e;
      Smask += me;
      const float* rv = comb + (long)(b * T + j) * 192 + 128;
      o0 += me * rv[2 * h];
      o1 += me * rv[2 * h + 1];
    }
    Sall = wred_sum(Sall);
    Smask = wred_sum(Smask);
    o0 = wred_sum(o0);
    o1 = wred_sum(o1);
    if (lane == 0) {
      float S2 = Smask / Sall;                 // sum of softmax*mask
      float coef = 1.0f / (Sall * (S2 + 1e-13f));
      long base = (long)(b * T + i) * 64 + 2 * h;
      atto[base] = o0 * coef;
      atto[base + 1] = o1 * coef;
    }
  }
}

// ---------------------------------------------------------------------------
// L2 row norm over E=300, one wave per row
// ---------------------------------------------------------------------------
__global__ void rownorm_kernel(const float* __restrict__ x, float* __restrict__ out) {
  int row = blockIdx.x;
  int lane = threadIdx.x;
  const float* p = x + (long)row * 300;
  float s = 0.0f;
  for (int idx = lane; idx < 300; idx += 32) { float v = p[idx]; s += v * v; }
  s = wred_sum(s);
  if (lane == 0) out[row] = sqrtf(s);
}

// ---------------------------------------------------------------------------
// Fused convs (k=1..5, bottom-right zero pad, channel-max for k>1) + Gaussian
// kernel pooling over D. combined: (B,2,Q,D). pkq: (B,5,Q,11).
// ---------------------------------------------------------------------------
DEVFN float conv_max(const float* __restrict__ wgt, const float* __restrict__ bias,
                     const float x[2][5][5], int k) {
  float best = -3.4e38f;
  for (int co = 0; co < 16; ++co) {
    float s = bias[co];
    for (int ci = 0; ci < 2; ++ci)
      for (int a = 0; a < k; ++a)
        for (int e = 0; e < k; ++e)
          s += wgt[((co * 2 + ci) * k + a) * k + e] * x[ci][a][e];
    best = fmaxf(best, s);
  }
  return best;
}

__global__ void convpool_kernel(const float* __restrict__ comb2,
                                const float* __restrict__ qm, const float* __restrict__ dm,
                                const float* __restrict__ c1w, const float* __restrict__ c1b,
                                const float* __restrict__ c2w, const float* __restrict__ c2b,
                                const float* __restrict__ c3w, const float* __restrict__ c3b,
                                const float* __restrict__ c4w, const float* __restrict__ c4b,
                                const float* __restrict__ c5w, const float* __restrict__ c5b,
                                float* __restrict__ pkq) {
  const int Q = 32, D = 512;
  const float MUv[11] = {1.0f, 0.9f, 0.7f, 0.5f, 0.3f, 0.1f,
                         -0.1f, -0.3f, -0.5f, -0.7f, -0.9f};
  const float SIGv[11] = {1e-4f, 0.1f, 0.1f, 0.1f, 0.1f, 0.1f,
                          0.1f, 0.1f, 0.1f, 0.1f, 0.1f};
  int b = blockIdx.y, q = blockIdx.x;
  int tid = threadIdx.x; // 256
  float acc[5][11];
  #pragma unroll
  for (int c = 0; c < 5; ++c)
    #pragma unroll
    for (int k = 0; k < 11; ++k) acc[c][k] = 0.0f;
  float qmv = qm[b * Q + q];

  for (int d = tid; d < D; d += 256) {
    float w[2][5][5];
    #pragma unroll
    for (int ci = 0; ci < 2; ++ci)
      #pragma unroll
      for (int dq = 0; dq < 5; ++dq)
        #pragma unroll
        for (int dd = 0; dd < 5; ++dd) {
          int qq = q + dq, dx = d + dd;
          float sc = (qq < Q && dx < D) ? 1.0f : 0.0f;   // branchless zero-pad
          int qc = (qq < Q) ? qq : (Q - 1);
          int dc = (dx < D) ? dx : (D - 1);
          w[ci][dq][dd] = comb2[((b * 2 + ci) * Q + qc) * D + dc] * sc;
        }
    float y[5];
    y[0] = c1w[0] * w[0][0][0] + c1w[1] * w[1][0][0] + c1b[0];
    y[1] = conv_max(c2w, c2b, w, 2);
    y[2] = conv_max(c3w, c3b, w, 3);
    y[3] = conv_max(c4w, c4b, w, 4);
    y[4] = conv_max(c5w, c5b, w, 5);
    float mprod = qmv * dm[b * D + d];
    #pragma unroll
    for (int c = 0; c < 5; ++c) {
      float val = y[c];
      #pragma unroll
      for (int k = 0; k < 11; ++k) {
        float diff = val - MUv[k];
        acc[c][k] += expf(-diff * diff / (2.0f * SIGv[k] * SIGv[k])) * mprod;
      }
    }
  }

  // deterministic reduction: wave shfl -> LDS across 8 waves
  __shared__ float sred[8][55];
  int lane = tid & 31, wv = tid >> 5;
  #pragma unroll
  for (int c = 0; c < 5; ++c)
    #pragma unroll
    for (int k = 0; k < 11; ++k) {
      float v = wred_sum(acc[c][k]);
      if (lane == 0) sred[wv][c * 11 + k] = v;
    }
  __syncthreads();
  if (tid < 55) {
    float s = 0.0f;
    for (int i = 0; i < 8; ++i) s += sred[i][tid];
    int c = tid / 11, k = tid % 11;
    pkq[((b * 5 + c) * Q + q) * 11 + k] = s;
  }
}

// ---------------------------------------------------------------------------
// Final: log-pooling over Q, two dense(55->1), combine(2->1). One block per b.
// ---------------------------------------------------------------------------
__global__ void final_kernel(const float* __restrict__ pkq,
                             const float* __restrict__ qm, const float* __restrict__ dm,
                             const float* __restrict__ dw, const float* __restrict__ db,
                             const float* __restrict__ dmw, const float* __restrict__ dmb,
                             const float* __restrict__ cw, float* __restrict__ out) {
  const int Q = 32, D = 512;
  int b = blockIdx.x;
  int tid = threadIdx.x; // 64
  __shared__ float s1[64], s2[64];
  __shared__ float sdl;
  if (tid == 0) {
    float dl = 0.0f;
    for (int d = 0; d < D; ++d) dl += dm[b * D + d];
    sdl = dl;
  }
  __syncthreads();
  float dl = sdl;
  float p1 = 0.0f, p2 = 0.0f;
  if (tid < 55) {
    int c = tid / 11, k = tid % 11;
    float pk = 0.0f, pkm = 0.0f;
    for (int q = 0; q < Q; ++q) {
      float v = pkq[((b * 5 + c) * Q + q) * 11 + k];
      float mv = qm[b * Q + q];
      pk  += logf(fmaxf(v, 1e-10f)) * mv;
      pkm += logf(fmaxf(v / dl, 1e-10f)) * mv;
    }
    p1 = pk * dw[tid];
    p2 = pkm * dmw[tid];
  }
  s1[tid] = p1;
  s2[tid] = p2;
  __syncthreads();
  if (tid == 0) {
    float d1 = db[0], d2 = dmb[0];
    for (int t = 0; t < 55; ++t) { d1 += s1[t]; d2 += s2[t]; }
    out[b] = d1 * cw[0] + d2 * cw[1];
  }
}

// ---------------------------------------------------------------------------
// Orchestration
// ---------------------------------------------------------------------------
extern "C" void kernel_launch(void* const* d_in, const int* in_sizes, int n_in,
                              void* d_out, int out_size, void* d_ws, size_t ws_size,
                              hipStream_t stream) {
  (void)in_sizes; (void)n_in; (void)out_size; (void)ws_size;
  const float* qemb = (const float*)d_in[0];
  const float* demb = (const float*)d_in[1];
  const float* qm   = (const float*)d_in[2];
  const float* dm   = (const float*)d_in[3];
  const float* W1   = (const float*)d_in[6];
  const float* b1   = (const float*)d_in[7];
  const float* W2   = (const float*)d_in[8];
  const float* b2   = (const float*)d_in[9];
  const float* ffg  = (const float*)d_in[10];
  const float* ffb  = (const float*)d_in[11];
  const float* Wc   = (const float*)d_in[12];
  const float* bc   = (const float*)d_in[13];
  const float* Wo   = (const float*)d_in[14];
  const float* bo   = (const float*)d_in[15];
  const float* lng  = (const float*)d_in[16];
  const float* lnb  = (const float*)d_in[17];
  const float* c1w  = (const float*)d_in[18];
  const float* c1b  = (const float*)d_in[19];
  const float* c2w  = (const float*)d_in[20];
  const float* c2b  = (const float*)d_in[21];
  const float* c3w  = (const float*)d_in[22];
  const float* c3b  = (const float*)d_in[23];
  const float* c4w  = (const float*)d_in[24];
  const float* c4b  = (const float*)d_in[25];
  const float* c5w  = (const float*)d_in[26];
  const float* c5b  = (const float*)d_in[27];
  const float* dw   = (const float*)d_in[28];
  const float* db   = (const float*)d_in[29];
  const float* dmw  = (const float*)d_in[30];
  const float* dmb  = (const float*)d_in[31];
  const float* cw   = (const float*)d_in[32];

  const int Q = 32, D = 512, E = 300;
  float* ws = (float*)d_ws;
  // workspace layout (floats)
  float* dA = ws;                  // 8192*300 : encin_d / comb_d / pkq
  float* dB = dA + 8192 * 300;     // 8192*100 : h1_d / atto_d / combined
  float* dC = dB + 8192 * 100;     // 8192*300 : ff0_d / attlin_d
  float* dD = dC + 8192 * 300;     // 8192*300 : ff_d
  float* dE = dD + 8192 * 300;     // 8192*300 : dctx
  float* qA = dE + 8192 * 300;     // 512*300
  float* qB = qA + 512 * 300;      // 512*100
  float* qC = qB + 512 * 100;      // 512*300
  float* qD = qC + 512 * 300;      // 512*300
  float* qE = qD + 512 * 300;      // 512*300 : qctx
  float* qnorm = qE + 512 * 300;   // 512
  float* dnorm = qnorm + 512;      // 8192

  // ---- document encoder (BT = 8192) ----
  posadd_kernel<<<(8192 * 300 + 255) / 256, 256, 0, stream>>>(demb, dA, 512, 8192 * 300);
  gemm_wmma_f32<<<dim3(7, 512, 1), 32, 0, stream>>>(dA, 300, 0, W1, 100, 0, 0, b1,
      dB, 100, 0, 8192, 100, 300, 1, nullptr, 0, nullptr, 0);
  gemm_wmma_f32<<<dim3(19, 512, 1), 32, 0, stream>>>(dB, 100, 0, W2, 300, 0, 0, b2,
      dC, 300, 0, 8192, 300, 100, 0, nullptr, 0, nullptr, 0);
  ln_kernel<<<2048, 128, 0, stream>>>(dC, dA, ffg, ffb, nullptr, dD, 8192);
  gemm_wmma_f32<<<dim3(12, 512, 1), 32, 0, stream>>>(dD, 300, 0, Wc, 192, 0, 0, bc,
      dA, 192, 0, 8192, 192, 300, 0, nullptr, 0, nullptr, 0);
  attn_kernel<<<dim3(128, 16), 128, 0, stream>>>(dA, dm, dB, 512);
  gemm_wmma_f32<<<dim3(19, 512, 1), 32, 0, stream>>>(dB, 64, 0, Wo, 300, 0, 0, bo,
      dC, 300, 0, 8192, 300, 64, 0, nullptr, 0, nullptr, 0);
  ln_kernel<<<2048, 128, 0, stream>>>(dC, dD, lng, lnb, dm, dE, 8192);

  // ---- query encoder (BT = 512) ----
  posadd_kernel<<<(512 * 300 + 255) / 256, 256, 0, stream>>>(qemb, qA, 32, 512 * 300);
  gemm_wmma_f32<<<dim3(7, 32, 1), 32, 0, stream>>>(qA, 300, 0, W1, 100, 0, 0, b1,
      qB, 100, 0, 512, 100, 300, 1, nullptr, 0, nullptr, 0);
  gemm_wmma_f32<<<dim3(19, 32, 1), 32, 0, stream>>>(qB, 100, 0, W2, 300, 0, 0, b2,
      qC, 300, 0, 512, 300, 100, 0, nullptr, 0, nullptr, 0);
  ln_kernel<<<128, 128, 0, stream>>>(qC, qA, ffg, ffb, nullptr, qD, 512);
  gemm_wmma_f32<<<dim3(12, 32, 1), 32, 0, stream>>>(qD, 300, 0, Wc, 192, 0, 0, bc,
      qA, 192, 0, 512, 192, 300, 0, nullptr, 0, nullptr, 0);
  attn_kernel<<<dim3(8, 16), 128, 0, stream>>>(qA, qm, qB, 32);
  gemm_wmma_f32<<<dim3(19, 32, 1), 32, 0, stream>>>(qB, 64, 0, Wo, 300, 0, 0, bo,
      qC, 300, 0, 512, 300, 64, 0, nullptr, 0, nullptr, 0);
  ln_kernel<<<128, 128, 0, stream>>>(qC, qD, lng, lnb, qm, qE, 512);

  // ---- similarity planes: combined (B,2,Q,D), ch0 = tanh(cos), ch1 = tanh(dot)
  rownorm_kernel<<<512, 32, 0, stream>>>(qemb, qnorm);
  rownorm_kernel<<<8192, 32, 0, stream>>>(demb, dnorm);
  float* combined = dB; // 2*Q*D per batch
  gemm_wmma_f32<<<dim3(32, 2, 16), 32, 0, stream>>>(
      qemb, E, Q * E, demb, E, D * E, 1, nullptr,
      combined, D, 2 * Q * D, Q, D, E, 2, qnorm, Q, dnorm, D);
  gemm_wmma_f32<<<dim3(32, 2, 16), 32, 0, stream>>>(
      qE, E, Q * E, dE, E, D * E, 1, nullptr,
      combined + Q * D, D, 2 * Q * D, Q, D, E, 2, nullptr, 0, nullptr, 0);

  // ---- conv + kernel pooling ----
  float* pkq = dA;
  convpool_kernel<<<dim3(32, 16), 256, 0, stream>>>(combined, qm, dm,
      c1w, c1b, c2w, c2b, c3w, c3b, c4w, c4b, c5w, c5b, pkq);

  // ---- final dense ----
  final_kernel<<<16, 64, 0, stream>>>(pkq, qm, dm, dw, db, dmw, dmb, cw, (float*)d_out);
}